// FuzzyAttention_75325136437361
// MI455X (gfx1250) — compile-verified
//
#include <hip/hip_runtime.h>
#include <stdint.h>

typedef __attribute__((ext_vector_type(2))) float        v2f;
typedef __attribute__((ext_vector_type(8))) float        v8f;
typedef __attribute__((ext_vector_type(4))) unsigned int v4u;
typedef __attribute__((ext_vector_type(4))) int          v4i;
typedef __attribute__((ext_vector_type(8))) int          v8i;

#ifndef __has_builtin
#define __has_builtin(x) 0
#endif
#if __has_builtin(__builtin_amdgcn_tensor_load_to_lds) && __has_builtin(__builtin_amdgcn_s_wait_tensorcnt)
#define HAVE_TDM 1
#else
#define HAVE_TDM 0
#endif

namespace {

constexpr int B_ = 8, P_ = 256, C_ = 512, R_ = 64, M_ = 256;
constexpr int NROW = B_ * C_;          // 4096 (b,c) rows
constexpr int TR   = 16;               // rows per workgroup tile
constexpr int LDA  = 17;               // padded LDS line stride (floats) -> conflict-free
constexpr float EPS_  = 1.17549435e-38f;
// tau = 0.9 * (1/64)^(1/256)
constexpr float TAU_  = 0.88549704f;

__device__ __forceinline__ v8f wmma4(v2f a, v2f b, v8f c) {
  // D = A(16x4 f32) * B(4x16 f32) + C(16x16 f32)  -> v_wmma_f32_16x16x4_f32
  return __builtin_amdgcn_wmma_f32_16x16x4_f32(false, a, false, b, (short)0, c,
                                               false, false);
}

__device__ __forceinline__ v8f v8f_zero() {
  v8f z = {0.f, 0.f, 0.f, 0.f, 0.f, 0.f, 0.f, 0.f};
  return z;
}

// ---------------------------------------------------------------------------
// Software-pipelined K=256 GEMM accumulate for two 16-wide N tiles.
// A operand comes from LDS (layout XL[k*LDA + row]); B operands stream from
// global (L2-resident weight panels). Double-buffered register chunks of
// 8 K-steps (32 K values): loads of chunk c+1 are in flight while the 16
// WMMAs of chunk c execute, so waits become per-chunk, not per-WMMA.
// ---------------------------------------------------------------------------
__device__ __forceinline__ void gemm256(const float* __restrict__ XLb,
                                        const float* __restrict__ w0,
                                        const float* __restrict__ w1,
                                        int rowA, int kOff,
                                        v8f& acc0, v8f& acc1) {
  v2f a[2][8], b0[2][8], b1[2][8];
#pragma unroll
  for (int j = 0; j < 8; ++j) {            // prologue: chunk 0
    const int k = 4 * j;
    a[0][j].x = XLb[(k + kOff) * LDA + rowA];
    a[0][j].y = XLb[(k + kOff + 1) * LDA + rowA];
    b0[0][j] = *(const v2f*)(w0 + k);
    b1[0][j] = *(const v2f*)(w1 + k);
  }
#pragma unroll
  for (int c = 0; c < 8; ++c) {
    const int cur = c & 1, nxt = cur ^ 1;
    if (c < 7) {
#pragma unroll
      for (int j = 0; j < 8; ++j) {        // issue loads for chunk c+1
        const int k = 32 * (c + 1) + 4 * j;
        a[nxt][j].x = XLb[(k + kOff) * LDA + rowA];
        a[nxt][j].y = XLb[(k + kOff + 1) * LDA + rowA];
        b0[nxt][j] = *(const v2f*)(w0 + k);
        b1[nxt][j] = *(const v2f*)(w1 + k);
      }
    }
#pragma unroll
    for (int j = 0; j < 8; ++j) {          // consume chunk c
      acc0 = wmma4(a[cur][j], b0[cur][j], acc0);
      acc1 = wmma4(a[cur][j], b1[cur][j], acc1);
    }
  }
}

// ---------------------------------------------------------------------------
// Fused forward kernel: one workgroup = 16 rows (one b, 16 consecutive c).
// 256 threads = 8 waves; each wave owns 2 of the 16 N-column tiles (16 wide).
// ---------------------------------------------------------------------------
__global__ __launch_bounds__(256)
void fuzzy_main(const float* __restrict__ x,    // [B,P,C]
                const float* __restrict__ Wq, const float* __restrict__ bq,
                const float* __restrict__ Wk, const float* __restrict__ bk,
                const float* __restrict__ Wv, const float* __restrict__ bv,
                const float* __restrict__ Wo, const float* __restrict__ bo,
                const float* __restrict__ sigma,                // [R,M]
                float* __restrict__ out)        // [B,P,C] (+ loss elsewhere)
{
  // LDS plan (floats):
  //   XL  [256 k][17]  : X tile, layout XL[k*17 + row]  (also reused as OUT stage)
  //   QL  [256 m][17]  : q tile  (also reused as PRED stage)
  //   FIRE[16][64], WTS[16][64], FP[8 waves][16 rows]
  __shared__ float smem[256 * LDA + 256 * LDA + TR * R_ + TR * R_ + 8 * TR];
  float* XL   = smem;
  float* QL   = smem + 256 * LDA;
  float* FIRE = QL + 256 * LDA;
  float* WTS  = FIRE + TR * R_;
  float* FP   = WTS + TR * R_;

  const int tid  = threadIdx.x;
  const int lane = tid & 31;
  const int wave = tid >> 5;
  const int row0 = blockIdx.x * TR;           // global row = b*C + c
  const int b    = row0 / C_;
  const int c0   = row0 % C_;                 // 16-aligned, never crosses b

  // ---- Stage X tile into LDS: XL[p][c] = input[b, p, c0+c] -------------------
#if HAVE_TDM
  if (tid < 32) {
    const uint64_t gaddr = (uint64_t)(uintptr_t)(x + ((size_t)b * P_) * C_ + c0);
    const uint32_t ldsoff = (uint32_t)(uintptr_t)XL;   // low 32b of flat = LDS byte addr
    v4u g0;
    g0[0] = 1u;                                        // count=1, user descriptor
    g0[1] = ldsoff;                                    // lds_addr (bytes)
    g0[2] = (uint32_t)(gaddr & 0xFFFFFFFFu);           // global_addr[31:0]
    g0[3] = (uint32_t)((gaddr >> 32) & 0x01FFFFFFu) | (2u << 30);  // [56:32] + type=2
    // group1: data_size=4B(code2), pad_enable, pad_interval=16dw(code3), pad=1dw(code0)
    const uint32_t d0   = (2u << 16) | (1u << 20) | (3u << 22);
    const uint32_t td0  = (uint32_t)C_;     // tensor dim0 (contiguous c)
    const uint32_t td1  = (uint32_t)P_;     // tensor dim1 (p lines)
    const uint32_t t0   = 16u;              // tile_dim0: 16 c values
    const uint32_t t1   = (uint32_t)P_;     // tile_dim1: 256 p lines
    const uint64_t s0   = (uint64_t)C_;     // dim0 stride: step p => +C elements
    v8i g1;
    g1[0] = (int)d0;
    g1[1] = (int)((td0 & 0xFFFFu) << 16);                   // td0[15:0] @ bits63:48
    g1[2] = (int)((td0 >> 16) | ((td1 & 0xFFFFu) << 16));
    g1[3] = (int)((td1 >> 16) | (t0 << 16));
    g1[4] = (int)(t1 | (0u << 16));                         // tile_dim2 = 0 (2D)
    g1[5] = (int)(uint32_t)(s0 & 0xFFFFFFFFu);
    g1[6] = (int)(uint32_t)((s0 >> 32) & 0xFFFFu);          // dim1 stride unused (2D)
    g1[7] = 0;
    v4i zz4 = {0, 0, 0, 0};
#if __clang_major__ >= 23
    v8i zz8 = {0, 0, 0, 0, 0, 0, 0, 0};
    __builtin_amdgcn_tensor_load_to_lds(g0, g1, zz4, zz4, zz8, 0);
#else
    __builtin_amdgcn_tensor_load_to_lds(g0, g1, zz4, zz4, 0);
#endif
    __builtin_amdgcn_s_wait_tensorcnt(0);
  }
#else
  for (int idx = tid; idx < P_ * TR; idx += 256) {
    const int p = idx >> 4, cj = idx & 15;
    XL[p * LDA + cj] = x[((size_t)b * P_ + p) * C_ + c0 + cj];
  }
#endif
  __syncthreads();

  const int rowA = lane & 15;                 // A-operand row (M of 16x4 tile)
  const int kOff = (lane >> 4) << 1;          // lanes 16..31 hold K+2,K+3
  const int nc   = lane & 15;                 // B/C/D column within 16-wide tile
  const int rhal = (lane >> 4) << 3;          // D rows: VGPR i -> row i or i+8
  const int n0   = wave * 32;                 // this wave's first column

  // ---- q = X @ Wq^T + bq -----------------------------------------------------
  {
    v8f a0 = v8f_zero(), a1 = v8f_zero();
    gemm256(XL, Wq + (size_t)(n0 + nc) * P_ + kOff,
            Wq + (size_t)(n0 + 16 + nc) * P_ + kOff, rowA, kOff, a0, a1);
    const float q0 = bq[n0 + nc], q1 = bq[n0 + 16 + nc];
#pragma unroll
    for (int i = 0; i < 8; ++i) {
      const int rw = i + rhal;
      QL[(n0 + nc) * LDA + rw]      = a0[i] + q0;
      QL[(n0 + 16 + nc) * LDA + rw] = a1[i] + q1;
    }
  }
  __syncthreads();

  // ---- Pass 1: firing strengths over all rules -------------------------------
  const float inv2M = 0.5f / (float)M_;
  for (int r = 0; r < R_; ++r) {
    const float* Wkr = Wk + (size_t)r * M_ * P_;
    if (r + 1 < R_)
      __builtin_prefetch(Wkr + (size_t)M_ * P_ + (size_t)(n0 + nc) * P_, 0, 1);

    v8f k0 = v8f_zero(), k1 = v8f_zero();
    gemm256(XL, Wkr + (size_t)(n0 + nc) * P_ + kOff,
            Wkr + (size_t)(n0 + 16 + nc) * P_ + kOff, rowA, kOff, k0, k1);

    const float is0 = 1.f / (sigma[r * M_ + n0 + nc] + EPS_);
    const float is1 = 1.f / (sigma[r * M_ + n0 + 16 + nc] + EPS_);
    const float kb0 = bk[r * M_ + n0 + nc];
    const float kb1 = bk[r * M_ + n0 + 16 + nc];
#pragma unroll
    for (int i = 0; i < 8; ++i) {
      const int rw = i + rhal;
      const float d0 = (QL[(n0 + nc) * LDA + rw]      - (k0[i] + kb0)) * is0;
      const float d1 = (QL[(n0 + 16 + nc) * LDA + rw] - (k1[i] + kb1)) * is1;
      float s = d0 * d0 + d1 * d1;
      // reduce over the 16 lanes of each half (columns of this wave)
      s += __shfl_xor(s, 1, 32);
      s += __shfl_xor(s, 2, 32);
      s += __shfl_xor(s, 4, 32);
      s += __shfl_xor(s, 8, 32);
      if ((lane & 15) == 0) FP[wave * TR + rw] = s;   // fixed slot per (wave,row)
    }
    __syncthreads();
    if (tid < TR) {
      float s = 0.f;
#pragma unroll
      for (int w = 0; w < 8; ++w) s += FP[w * TR + tid];   // deterministic order
      FIRE[tid * R_ + r] = __expf(-inv2M * s) + EPS_;
    }
    __syncthreads();
  }

  // ---- Per-row normalize + nucleus top-p mask + renormalize ------------------
  if (tid < TR) {
    float s[R_], so[R_];
    float sum = 0.f;
    for (int r = 0; r < R_; ++r) { s[r] = FIRE[tid * R_ + r]; sum += s[r]; }
    const float inv = 1.f / (sum + EPS_);
    for (int r = 0; r < R_; ++r) { s[r] *= inv; so[r] = s[r]; }
    // insertion sort, descending
    for (int i = 1; i < R_; ++i) {
      const float key = so[i];
      int j = i - 1;
      while (j >= 0 && so[j] < key) { so[j + 1] = so[j]; --j; }
      so[j + 1] = key;
    }
    float cum = 0.f, best = 3.402823466e38f;
    int bi = 0;
    for (int r = 0; r < R_; ++r) {
      cum += so[r];
      float e = cum - TAU_;
      if (e < 0.f) e = 1.0f;
      if (e < best) { best = e; bi = r; }       // first-min like argmin
    }
    const float thresh = so[bi];
    float msum = 0.f;
    for (int r = 0; r < R_; ++r) {
      const float wv = (s[r] >= thresh) ? s[r] : 0.f;
      WTS[tid * R_ + r] = wv;
      msum += wv;
    }
    const float inv2 = 1.f / (msum + EPS_);
    for (int r = 0; r < R_; ++r) WTS[tid * R_ + r] *= inv2;
  }
  __syncthreads();

  // ---- Pass 2: pred = sum_r w[row,r] * (X @ Wv_r^T + bv_r) -------------------
  v8f p0 = v8f_zero(), p1 = v8f_zero();
  for (int r = 0; r < R_; ++r) {
    const float* Wvr = Wv + (size_t)r * M_ * P_;
    if (r + 1 < R_)
      __builtin_prefetch(Wvr + (size_t)M_ * P_ + (size_t)(n0 + nc) * P_, 0, 1);

    v8f u0 = v8f_zero(), u1 = v8f_zero();
    gemm256(XL, Wvr + (size_t)(n0 + nc) * P_ + kOff,
            Wvr + (size_t)(n0 + 16 + nc) * P_ + kOff, rowA, kOff, u0, u1);

    const float vb0 = bv[r * M_ + n0 + nc];
    const float vb1 = bv[r * M_ + n0 + 16 + nc];
#pragma unroll
    for (int i = 0; i < 8; ++i) {
      const int rw = i + rhal;
      const float w = WTS[rw * R_ + r];
      p0[i] = fmaf(w, u0[i] + vb0, p0[i]);
      p1[i] = fmaf(w, u1[i] + vb1, p1[i]);
    }
  }
  // stage pred into QL[m][row] (q no longer needed)
#pragma unroll
  for (int i = 0; i < 8; ++i) {
    const int rw = i + rhal;
    QL[(n0 + nc) * LDA + rw]      = p0[i];
    QL[(n0 + 16 + nc) * LDA + rw] = p1[i];
  }
  __syncthreads();

  // ---- out = pred @ Wo^T + bo ------------------------------------------------
  {
    v8f o0 = v8f_zero(), o1 = v8f_zero();
    gemm256(QL, Wo + (size_t)(n0 + nc) * M_ + kOff,        // Wo[p, m]
            Wo + (size_t)(n0 + 16 + nc) * M_ + kOff, rowA, kOff, o0, o1);
    const float b0v = bo[n0 + nc], b1v = bo[n0 + 16 + nc];
#pragma unroll
    for (int i = 0; i < 8; ++i) {
      const int rw = i + rhal;
      XL[(n0 + nc) * LDA + rw]      = o0[i] + b0v;   // XL reused as [p][row] stage
      XL[(n0 + 16 + nc) * LDA + rw] = o1[i] + b1v;
    }
  }
  __syncthreads();

  // ---- coalesced store: out[b, p, c0..c0+15] ---------------------------------
  {
    const int p = tid;                        // 256 threads = 256 p lines
    float* dst = out + ((size_t)b * P_ + p) * C_ + c0;
    const float* src = XL + p * LDA;
#pragma unroll
    for (int j = 0; j < 16; j += 4) {
      float4 v;
      v.x = src[j]; v.y = src[j + 1]; v.z = src[j + 2]; v.w = src[j + 3];
      *(float4*)(dst + j) = v;
    }
  }
}

// ---------------------------------------------------------------------------
// Loss kernels (deterministic two-stage reductions through d_ws)
// ---------------------------------------------------------------------------
__global__ __launch_bounds__(256)
void wq_wo_loss(const float* __restrict__ Wq, const float* __restrict__ Wo,
                float* __restrict__ part /* [256] */) {
  __shared__ float red[256];
  const int m = blockIdx.x, m2 = threadIdx.x;
  float dot = 0.f;
  for (int p = 0; p < P_; ++p)
    dot = fmaf(Wq[(size_t)m * P_ + p], Wo[(size_t)p * M_ + m2], dot);
  const float d = dot - ((m == m2) ? 1.f : 0.f);
  red[m2] = d * d;
  __syncthreads();
  for (int s = 128; s > 0; s >>= 1) {
    if (m2 < s) red[m2] += red[m2 + s];
    __syncthreads();
  }
  if (m2 == 0) part[m] = red[0];
}

__global__ __launch_bounds__(256)
void l1_loss(const float* __restrict__ Wk, const float* __restrict__ bk,
             const float* __restrict__ Wv, const float* __restrict__ bv,
             const float* __restrict__ bq, const float* __restrict__ bo,
             float* __restrict__ part /* [grid][4] */) {
  const size_t NW = (size_t)R_ * M_ * P_;   // 4,194,304
  const size_t NB = (size_t)R_ * M_;        // 16,384
  const size_t stride = (size_t)gridDim.x * blockDim.x;
  float kl = 0.f, vl = 0.f, q2 = 0.f, o2 = 0.f;
  for (size_t i = (size_t)blockIdx.x * blockDim.x + threadIdx.x; i < NW; i += stride) {
    kl += __builtin_fabsf(Wk[i]);
    vl += __builtin_fabsf(Wv[i]);
  }
  for (size_t i = (size_t)blockIdx.x * blockDim.x + threadIdx.x; i < NB; i += stride) {
    kl += __builtin_fabsf(bk[i]);
    vl += __builtin_fabsf(bv[i]);
  }
  for (size_t i = (size_t)blockIdx.x * blockDim.x + threadIdx.x; i < (size_t)M_; i += stride) {
    q2 = fmaf(bq[i], bq[i], q2);
  }
  for (size_t i = (size_t)blockIdx.x * blockDim.x + threadIdx.x; i < (size_t)P_; i += stride) {
    o2 = fmaf(bo[i], bo[i], o2);
  }
  __shared__ float4 red[256];
  float4 v;
  v.x = kl; v.y = vl; v.z = q2; v.w = o2;
  red[threadIdx.x] = v;
  __syncthreads();
  for (int s = 128; s > 0; s >>= 1) {
    if ((int)threadIdx.x < s) {
      float4 a = red[threadIdx.x], b = red[threadIdx.x + s];
      a.x += b.x; a.y += b.y; a.z += b.z; a.w += b.w;
      red[threadIdx.x] = a;
    }
    __syncthreads();
  }
  if (threadIdx.x == 0) {
    float4 a = red[0];
    part[(size_t)blockIdx.x * 4 + 0] = a.x;
    part[(size_t)blockIdx.x * 4 + 1] = a.y;
    part[(size_t)blockIdx.x * 4 + 2] = a.z;
    part[(size_t)blockIdx.x * 4 + 3] = a.w;
  }
}

__global__ void loss_final(const float* __restrict__ wpart,
                           const float* __restrict__ lpart, int nl,
                           float* __restrict__ out_loss) {
  if (threadIdx.x == 0 && blockIdx.x == 0) {
    float wsum = 0.f;
    for (int i = 0; i < 256; ++i) wsum += wpart[i];
    float kl = 0.f, vl = 0.f, q2 = 0.f, o2 = 0.f;
    for (int i = 0; i < nl; ++i) {
      kl += lpart[i * 4 + 0];
      vl += lpart[i * 4 + 1];
      q2 += lpart[i * 4 + 2];
      o2 += lpart[i * 4 + 3];
    }
    const float loss = 0.01f * (__fsqrt_rn(wsum) + __fsqrt_rn(q2) + __fsqrt_rn(o2)) +
                       0.001f * (kl + vl);
    *out_loss = loss;
  }
}

}  // namespace

extern "C" void kernel_launch(void* const* d_in, const int* in_sizes, int n_in,
                              void* d_out, int out_size, void* d_ws, size_t ws_size,
                              hipStream_t stream) {
  (void)in_sizes; (void)n_in; (void)out_size; (void)ws_size;
  const float* x     = (const float*)d_in[0];
  const float* Wq    = (const float*)d_in[1];
  const float* bq    = (const float*)d_in[2];
  const float* Wk    = (const float*)d_in[3];
  const float* bk    = (const float*)d_in[4];
  const float* Wv    = (const float*)d_in[5];
  const float* bv    = (const float*)d_in[6];
  const float* Wo    = (const float*)d_in[7];
  const float* bo    = (const float*)d_in[8];
  const float* sigma = (const float*)d_in[9];

  float* out   = (float*)d_out;              // [8,256,512] flat, loss at +1048576
  float* wpart = (float*)d_ws;               // 256 floats
  float* lpart = wpart + 256;                // 128*4 floats
  constexpr int L1B = 128;

  fuzzy_main<<<dim3(NROW / TR), dim3(256), 0, stream>>>(
      x, Wq, bq, Wk, bk, Wv, bv, Wo, bo, sigma, out);
  wq_wo_loss<<<dim3(256), dim3(256), 0, stream>>>(Wq, Wo, wpart);
  l1_loss<<<dim3(L1B), dim3(256), 0, stream>>>(Wk, bk, Wv, bv, bq, bo, lpart);
  loss_final<<<dim3(1), dim3(64), 0, stream>>>(wpart, lpart, L1B,
                                               out + (size_t)B_ * P_ * C_);
}